// MultiHeadAttention_40570261078071
// MI455X (gfx1250) — compile-verified
//
#include <hip/hip_runtime.h>
#include <hip/hip_bf16.h>
#include <stdint.h>

typedef float v2f __attribute__((ext_vector_type(2)));
typedef float v8f __attribute__((ext_vector_type(8)));

#define WMMA_F32(a, b, c) \
    __builtin_amdgcn_wmma_f32_16x16x4_f32(false, (a), false, (b), (short)0, (c), false, false)

constexpr int L = 2048, B = 2, C = 1024, H = 16, HD = 64;
constexpr int M_ROWS = L * B;             // 4096
constexpr float NEGINF = -1e30f;
constexpr float SCALE  = 0.125f;          // HD^-0.5
constexpr float INV_H  = 1.0f / 16.0f;

constexpr int BP = 38;                    // B-tile LDS stride (transposed [n][k])
constexpr int VP = 70;                    // V-stage LDS stride (transposed [d][key])

// ---------------------------------------------------------------------------
// Kernel 0: zero the attn_avg region of d_out (we accumulate atomically).
// ---------------------------------------------------------------------------
__global__ void zero_kernel(float* __restrict__ p, size_t n) {
    size_t i = (size_t)blockIdx.x * blockDim.x + threadIdx.x;
    if (i < n) p[i] = 0.0f;
}

// ---------------------------------------------------------------------------
// Kernel 1: QKV projection.  qkv = x @ Wqkv + bqkv ; q *= scale.
// M=4096, K=1024, N=3072.  Block tile 64x128, 8 waves, 32x32 per wave.
// B tile stored transposed in LDS so b-frags are contiguous ds_load_b64.
// ---------------------------------------------------------------------------
__global__ __launch_bounds__(256)
void qkv_gemm_kernel(const float* __restrict__ x, const float* __restrict__ Wqkv,
                     const float* __restrict__ bqkv,
                     float* __restrict__ qws, float* __restrict__ kws,
                     float* __restrict__ vws)
{
    constexpr int N = 3 * C;              // 3072
    __shared__ float Als[64 * 36];        // [m][k], 64 x 32 pad 36
    __shared__ float Bls[128 * BP];       // [n][k], 128 x 32 pad 38 (transposed)

    const int tid  = threadIdx.x;
    const int wave = tid >> 5, lane = tid & 31;
    const int lm = lane & 15, hi = lane >> 4;
    const int wm = wave >> 2, wn = wave & 3;     // 2x4 wave grid
    const int m0 = blockIdx.y * 64;
    const int n0 = blockIdx.x * 128;

    v8f acc[2][2];
    for (int i = 0; i < 2; ++i)
        for (int j = 0; j < 2; ++j)
            for (int e = 0; e < 8; ++e) acc[i][j][e] = 0.0f;

    for (int k0 = 0; k0 < C; k0 += 32) {
        for (int i = 0; i < 8; ++i) {            // A: 64x32, coalesced
            int idx = tid + i * 256;
            int r = idx >> 5, cc = idx & 31;
            Als[r * 36 + cc] = x[(size_t)(m0 + r) * C + k0 + cc];
        }
        for (int i = 0; i < 16; ++i) {           // B: 32x128, store transposed
            int idx = tid + i * 256;
            int r = idx >> 7, cc = idx & 127;    // r=k, cc=n
            Bls[cc * BP + r] = Wqkv[(size_t)(k0 + r) * N + n0 + cc];
        }
        if (k0 + 32 < C) {                       // prefetch next tiles
            __builtin_prefetch(&x[(size_t)(m0 + (tid >> 5)) * C + k0 + 32 + (tid & 31)], 0, 0);
            __builtin_prefetch(&Wqkv[(size_t)(k0 + 32 + (tid >> 7)) * N + n0 + (tid & 127)], 0, 0);
        }
        __syncthreads();
        for (int kk = 0; kk < 32; kk += 4) {
            v2f afrag[2], bfrag[2];
            for (int mt = 0; mt < 2; ++mt) {
                const float* ap = &Als[(wm * 32 + mt * 16 + lm) * 36 + kk + hi * 2];
                afrag[mt][0] = ap[0]; afrag[mt][1] = ap[1];
            }
            for (int nt = 0; nt < 2; ++nt) {
                const float* bp = &Bls[(wn * 32 + nt * 16 + lm) * BP + kk + hi * 2];
                bfrag[nt][0] = bp[0]; bfrag[nt][1] = bp[1];
            }
            for (int mt = 0; mt < 2; ++mt)
                for (int nt = 0; nt < 2; ++nt)
                    acc[mt][nt] = WMMA_F32(afrag[mt], bfrag[nt], acc[mt][nt]);
        }
        __syncthreads();
    }

    for (int mt = 0; mt < 2; ++mt)
        for (int nt = 0; nt < 2; ++nt) {
            int gn = n0 + wn * 32 + nt * 16 + lm;
            int part = gn >> 10;                  // 0=q,1=k,2=v
            int hc = gn & 1023;
            int h = hc >> 6, dd = hc & 63;
            float bias = bqkv[gn];
            float* dst = (part == 0) ? qws : (part == 1 ? kws : vws);
            float mul = (part == 0) ? SCALE : 1.0f;
            for (int e = 0; e < 8; ++e) {
                int gm = m0 + wm * 32 + mt * 16 + e + hi * 8;  // row = l*B + b
                int l = gm >> 1, bb = gm & 1;
                dst[((size_t)(bb * H + h) * L + l) * HD + dd] =
                    (acc[mt][nt][e] + bias) * mul;
            }
        }
}

// ---------------------------------------------------------------------------
// Kernel 2: attention.  One WG per (head, 32-query tile).
// Full 32 x 2048 score tile kept in LDS (~300KB of the 320KB WGP LDS).
// ---------------------------------------------------------------------------
__global__ __launch_bounds__(256)
void attn_kernel(const float* __restrict__ qws, const float* __restrict__ kws,
                 const float* __restrict__ vws,
                 const uint8_t* __restrict__ amask,   // (L,L) bool, true = masked
                 const uint8_t* __restrict__ kpm,     // (B,L) bool, true = masked
                 float* __restrict__ attn_avg,        // (B,L,L)
                 float* __restrict__ ctx)             // (L,B,C)
{
    extern __shared__ float smem[];
    float* Sls = smem;                       // 32 x 2052 (padded)
    float* Qls = smem + 32 * 2052;           // 32 x 68
    float* Kst = Qls + 32 * 68;              // 8 waves x (16 x 68); >= 64*VP

    const int tid  = threadIdx.x;
    const int wave = tid >> 5, lane = tid & 31;
    const int lm = lane & 15, hi = lane >> 4;
    const int q0 = blockIdx.x * 32;
    const int bh = blockIdx.y;
    const int bb = bh >> 4;                  // batch
    const int h  = bh & 15;                  // head
    const float* qg = qws + (size_t)bh * L * HD;
    const float* kg = kws + (size_t)bh * L * HD;
    const float* vg = vws + (size_t)bh * L * HD;

    // ---- load Q tile (32 x 64), already scaled ----
    for (int i = 0; i < 8; ++i) {
        int idx = tid + i * 256;
        int r = idx >> 6, cc = idx & 63;
        Qls[r * 68 + cc] = qg[(size_t)(q0 + r) * HD + cc];
    }
    __syncthreads();

    // ---- phase A: S = Q K^T with masking, written to LDS ----
    float* kst = Kst + wave * (16 * 68);     // wave-private K staging
    for (int nt = wave; nt < L / 16; nt += 8) {
        if (nt + 8 < L / 16)                 // prefetch this wave's next K tile
            __builtin_prefetch(
                &kg[(size_t)((nt + 8) * 16 + (lane >> 1)) * HD + (lane & 1) * 32], 0, 0);
        for (int i = 0; i < 32; ++i) {       // stage K tile 16x64
            int idx = lane + i * 32;
            int r = idx >> 6, cc = idx & 63;
            kst[r * 68 + cc] = kg[(size_t)(nt * 16 + r) * HD + cc];
        }
        int key = nt * 16 + lm;
        bool kmask = (kpm[bb * L + key] != 0);
        for (int mt = 0; mt < 2; ++mt) {
            v8f acc;
            for (int e = 0; e < 8; ++e) acc[e] = 0.0f;
            for (int kk = 0; kk < HD; kk += 4) {
                v2f a, bf;
                const float* ap = &Qls[(mt * 16 + lm) * 68 + kk + hi * 2];
                a[0] = ap[0]; a[1] = ap[1];
                const float* bp = &kst[lm * 68 + kk + hi * 2];
                bf[0] = bp[0]; bf[1] = bp[1];
                acc = WMMA_F32(a, bf, acc);
            }
            for (int e = 0; e < 8; ++e) {
                int m = mt * 16 + e + hi * 8;
                int qrow = q0 + m;
                float s = acc[e];
                if (kmask || amask[(size_t)qrow * L + key]) s = NEGINF;
                Sls[m * 2052 + key] = s;
            }
        }
    }
    __syncthreads();

    // ---- phase B: softmax per row + head-averaged probability output ----
    for (int rr = 0; rr < 4; ++rr) {
        int m = wave * 4 + rr;
        float* row = Sls + m * 2052;
        float mx = -3.4e38f;
        for (int cc = lane; cc < L; cc += 32) mx = fmaxf(mx, row[cc]);
        for (int off = 16; off > 0; off >>= 1) mx = fmaxf(mx, __shfl_xor(mx, off));
        float sum = 0.0f;
        for (int cc = lane; cc < L; cc += 32) {
            float e = __expf(row[cc] - mx);
            row[cc] = e;
            sum += e;
        }
        for (int off = 16; off > 0; off >>= 1) sum += __shfl_xor(sum, off);
        float inv = 1.0f / sum;
        float* aa = attn_avg + (size_t)bb * L * L + (size_t)(q0 + m) * L;
        for (int cc = lane; cc < L; cc += 32) {
            float p = row[cc] * inv;
            row[cc] = p;
            atomicAdd(&aa[cc], p * INV_H);
        }
    }
    __syncthreads();

    // ---- phase C: O = P @ V, V staged transposed in LDS (reuse Kst) ----
    {
        float* Vst = Kst;                    // 64 x VP transposed [d][key]
        int mt = wave >> 2, vt = wave & 3;
        v8f acc;
        for (int e = 0; e < 8; ++e) acc[e] = 0.0f;
        for (int kc = 0; kc < L; kc += 64) {
            for (int i = 0; i < 16; ++i) {   // stage 64 keys x 64 d, coalesced
                int idx = tid + i * 256;
                int r = idx >> 6, cc = idx & 63;       // r=key, cc=d
                Vst[cc * VP + r] = vg[(size_t)(kc + r) * HD + cc];
            }
            __syncthreads();
            for (int kk = 0; kk < 64; kk += 4) {
                v2f a, bf;
                const float* ap = &Sls[(mt * 16 + lm) * 2052 + kc + kk + hi * 2];
                a[0] = ap[0]; a[1] = ap[1];
                const float* bp = &Vst[(vt * 16 + lm) * VP + kk + hi * 2];
                bf[0] = bp[0]; bf[1] = bp[1];
                acc = WMMA_F32(a, bf, acc);
            }
            __syncthreads();
        }
        for (int e = 0; e < 8; ++e) {
            int m = mt * 16 + e + hi * 8;
            int l = q0 + m;
            ctx[((size_t)l * B + bb) * C + h * HD + vt * 16 + lm] = acc[e];
        }
    }
}

// ---------------------------------------------------------------------------
// Kernel 3: output projection.  out = ctx @ Wout + bout.  M=4096,K=1024,N=1024.
// ---------------------------------------------------------------------------
__global__ __launch_bounds__(256)
void out_gemm_kernel(const float* __restrict__ ctx, const float* __restrict__ Wout,
                     const float* __restrict__ bout, float* __restrict__ out)
{
    __shared__ float Als[64 * 36];
    __shared__ float Bls[128 * BP];          // [n][k] transposed

    const int tid  = threadIdx.x;
    const int wave = tid >> 5, lane = tid & 31;
    const int lm = lane & 15, hi = lane >> 4;
    const int wm = wave >> 2, wn = wave & 3;
    const int m0 = blockIdx.y * 64;
    const int n0 = blockIdx.x * 128;

    v8f acc[2][2];
    for (int i = 0; i < 2; ++i)
        for (int j = 0; j < 2; ++j)
            for (int e = 0; e < 8; ++e) acc[i][j][e] = 0.0f;

    for (int k0 = 0; k0 < C; k0 += 32) {
        for (int i = 0; i < 8; ++i) {
            int idx = tid + i * 256;
            int r = idx >> 5, cc = idx & 31;
            Als[r * 36 + cc] = ctx[(size_t)(m0 + r) * C + k0 + cc];
        }
        for (int i = 0; i < 16; ++i) {
            int idx = tid + i * 256;
            int r = idx >> 7, cc = idx & 127;
            Bls[cc * BP + r] = Wout[(size_t)(k0 + r) * C + n0 + cc];
        }
        if (k0 + 32 < C) {
            __builtin_prefetch(&ctx[(size_t)(m0 + (tid >> 5)) * C + k0 + 32 + (tid & 31)], 0, 0);
            __builtin_prefetch(&Wout[(size_t)(k0 + 32 + (tid >> 7)) * C + n0 + (tid & 127)], 0, 0);
        }
        __syncthreads();
        for (int kk = 0; kk < 32; kk += 4) {
            v2f afrag[2], bfrag[2];
            for (int mt = 0; mt < 2; ++mt) {
                const float* ap = &Als[(wm * 32 + mt * 16 + lm) * 36 + kk + hi * 2];
                afrag[mt][0] = ap[0]; afrag[mt][1] = ap[1];
            }
            for (int nt = 0; nt < 2; ++nt) {
                const float* bp = &Bls[(wn * 32 + nt * 16 + lm) * BP + kk + hi * 2];
                bfrag[nt][0] = bp[0]; bfrag[nt][1] = bp[1];
            }
            for (int mt = 0; mt < 2; ++mt)
                for (int nt = 0; nt < 2; ++nt)
                    acc[mt][nt] = WMMA_F32(afrag[mt], bfrag[nt], acc[mt][nt]);
        }
        __syncthreads();
    }

    for (int mt = 0; mt < 2; ++mt)
        for (int nt = 0; nt < 2; ++nt) {
            int gn = n0 + wn * 32 + nt * 16 + lm;
            float bias = bout[gn];
            for (int e = 0; e < 8; ++e) {
                int gm = m0 + wm * 32 + mt * 16 + e + hi * 8;
                out[(size_t)gm * C + gn] = acc[mt][nt][e] + bias;
            }
        }
}

// ---------------------------------------------------------------------------
extern "C" void kernel_launch(void* const* d_in, const int* in_sizes, int n_in,
                              void* d_out, int out_size, void* d_ws, size_t ws_size,
                              hipStream_t stream)
{
    (void)in_sizes; (void)n_in; (void)out_size; (void)ws_size;

    const float*   x     = (const float*)d_in[0];
    const float*   Wqkv  = (const float*)d_in[1];
    const float*   bqkv  = (const float*)d_in[2];
    const float*   Wout  = (const float*)d_in[3];
    const float*   bout  = (const float*)d_in[4];
    const uint8_t* amask = (const uint8_t*)d_in[5];
    const uint8_t* kpm   = (const uint8_t*)d_in[6];

    float* out      = (float*)d_out;                       // (L,B,C)
    float* attn_avg = out + (size_t)L * B * C;             // (B,L,L)

    constexpr size_t HEADSZ = (size_t)B * H * L * HD;      // 4 Mi floats each
    float* qws = (float*)d_ws;
    float* kws = qws + HEADSZ;
    float* vws = kws + HEADSZ;
    float* ctx = vws + HEADSZ;                             // (L,B,C)

    // 0) zero attn_avg (atomic accumulation target)
    {
        size_t n = (size_t)B * L * L;
        zero_kernel<<<(unsigned)((n + 255) / 256), 256, 0, stream>>>(attn_avg, n);
    }
    // 1) QKV projection
    qkv_gemm_kernel<<<dim3(3 * C / 128, M_ROWS / 64), 256, 0, stream>>>(
        x, Wqkv, bqkv, qws, kws, vws);
    // 2) attention (dynamic LDS: scores 32x2052 + Q 32x68 + K/V stage 8x16x68)
    {
        size_t shmem = (size_t)(32 * 2052 + 32 * 68 + 8 * 16 * 68) * sizeof(float);
        attn_kernel<<<dim3(L / 32, B * H), 256, shmem, stream>>>(
            qws, kws, vws, amask, kpm, attn_avg, ctx);
    }
    // 3) output projection
    out_gemm_kernel<<<dim3(C / 128, M_ROWS / 64), 256, 0, stream>>>(
        ctx, Wout, bout, out);
}